// Shapelet_19370302505357
// MI455X (gfx1250) — compile-verified
//
#include <hip/hip_runtime.h>

// Problem constants (match reference)
#define B_ 8
#define C_ 3
#define T_ 384
#define N_ 16
#define L_ 24
#define STRIDE_ 2
#define M_ 181          // (T - L)/STRIDE + 1
#define GAMMA_ 0.05f
#define INVG_ 20.0f     // 1/GAMMA
#define BIGF 1e30f
#define BLK_ 192        // 6 wave32 waves; covers M=181 windows

// Numerically-stable soft-min of 3 values (matches -gamma*logsumexp(-x/gamma)).
__device__ __forceinline__ float softmin3(float a, float b, float c) {
    float m = fminf(a, fminf(b, c));
    float r = __expf((m - a) * INVG_) + __expf((m - b) * INVG_) + __expf((m - c) * INVG_);
    return fmaf(-GAMMA_, __logf(r), m);   // m - gamma*log(r), r in [1,3]
}

__global__ __launch_bounds__(BLK_) void shapelet_softdtw_kernel(
    const float* __restrict__ x,      // (B, C, T)
    const float* __restrict__ w,      // (N, C, L)
    float* __restrict__ out)          // [0:384) = out1 (B,N,C), [384:768) = out2
{
    const int tid = threadIdx.x;                  // window index m
    const int bid = blockIdx.x;                   // (b, n, c)
    const int c = bid % C_;
    const int n = (bid / C_) % N_;
    const int b = bid / (C_ * N_);

    __shared__ __align__(16) float lds_x[T_];     // one x row (b,c): 1536 B
    __shared__ __align__(16) float lds_s[L_ + 8]; // one shapelet row (n,c): 96 B
    __shared__ float red_min[BLK_];
    __shared__ float red_max[BLK_];

    const float* xrow = x + (size_t)(b * C_ + c) * T_;   // 1536-B aligned rows
    const float* srow = w + (size_t)(n * C_ + c) * L_;   // 96-B aligned rows

    // ---- gfx1250 async global->LDS staging, 16 B per lane (ASYNCcnt path) ----
    if (tid < (T_ / 4)) {                         // 96 lanes x b128 = 1536 B
        unsigned o = (unsigned)(size_t)&lds_x[0] + tid * 16u;
        const float* g = xrow + tid * 4;
        asm volatile("global_load_async_to_lds_b128 %0, %1, off"
                     :: "v"(o), "v"(g) : "memory");
    } else if (tid < (T_ / 4) + (L_ / 4)) {       // 6 lanes x b128 = 96 B
        const int t = tid - (T_ / 4);
        unsigned o = (unsigned)(size_t)&lds_s[0] + t * 16u;
        const float* g = srow + t * 4;
        asm volatile("global_load_async_to_lds_b128 %0, %1, off"
                     :: "v"(o), "v"(g) : "memory");
    }
    asm volatile("s_wait_asynccnt 0" ::: "memory");
    __syncthreads();

    // ---- per-lane 24x24 soft-DTW DP, fully in registers ----
    const int start = (tid < M_) ? tid * STRIDE_ : 0;   // clamp idle lanes to valid data
    float xw[L_];
#pragma unroll
    for (int l = 0; l < L_; ++l) xw[l] = lds_x[start + l];

    float prev[L_ + 1];
    prev[0] = 0.0f;
#pragma unroll
    for (int i = 1; i <= L_; ++i) prev[i] = BIGF;

    for (int j = 0; j < L_; ++j) {                // outer loop rolled (sj via LDS broadcast)
        const float sj = lds_s[j];
        float diag  = prev[0];                    // prev[i-1]
        float above = BIGF;                       // curr[0] = BIG
        prev[0] = BIGF;
#pragma unroll
        for (int i = 1; i <= L_; ++i) {           // inner loop unrolled: static prev[] indices
            const float diff = xw[i - 1] - sj;
            const float cur  = fmaf(diff, diff, softmin3(prev[i], above, diag));
            diag    = prev[i];
            prev[i] = cur;
            above   = cur;
        }
    }

    float d = prev[L_] * (1.0f / (float)L_);
    if (tid >= M_) d = BIGF;                      // idle lanes lose both reductions
    const float p = expf(-d * d);                 // EPS = 1

    // ---- block reduction over m: min(d), max(p) ----
    red_min[tid] = d;
    red_max[tid] = p;
    __syncthreads();
    for (int s = BLK_ / 2; s > 1; s >>= 1) {      // 96,48,24,12,6,3
        if (tid < s) {
            red_min[tid] = fminf(red_min[tid], red_min[tid + s]);
            red_max[tid] = fmaxf(red_max[tid], red_max[tid + s]);
        }
        __syncthreads();
    }
    if (tid == 0) {
        const float mind = fminf(fminf(red_min[0], red_min[1]), red_min[2]);
        const float maxp = fmaxf(fmaxf(red_max[0], red_max[1]), red_max[2]);
        const int oidx = (b * N_ + n) * C_ + c;   // (B,N,C) row-major
        out[oidx] = maxp;                         // out1
        out[(B_ * N_ * C_) + oidx] = mind;        // out2
    }
}

extern "C" void kernel_launch(void* const* d_in, const int* in_sizes, int n_in,
                              void* d_out, int out_size, void* d_ws, size_t ws_size,
                              hipStream_t stream) {
    (void)in_sizes; (void)n_in; (void)out_size; (void)d_ws; (void)ws_size;
    const float* x = (const float*)d_in[0];   // (8,3,384) f32
    const float* w = (const float*)d_in[1];   // (16,3,24) f32
    float* out = (float*)d_out;               // 768 f32
    dim3 grid(B_ * N_ * C_);                  // 384 blocks: (b,n,c)
    dim3 block(BLK_);                         // 192 threads = windows m
    hipLaunchKernelGGL(shapelet_softdtw_kernel, grid, block, 0, stream, x, w, out);
}